// GNNSeedAttention_42374147342558
// MI455X (gfx1250) — compile-verified
//
#include <hip/hip_runtime.h>
#include <hip/hip_bf16.h>
#include <math.h>

// ---------------------------------------------------------------------------
// CDNA5 (gfx1250, wave32) implementation of the GIN + seed-attention network.
// GEMMs use V_WMMA_F32_16X16X4_F32 (full fp32 matrix path).
// Block = 8 waves sharing one 64-col B panel staged in LDS (float2-paired so
// each B fragment is one ds_load_b64); each wave computes a 16x64 tile.
// ---------------------------------------------------------------------------

typedef float v2f __attribute__((ext_vector_type(2)));
typedef float v8f __attribute__((ext_vector_type(8)));

#define WAVE 32
#define WAVES_PER_BLOCK 8
#define ROWS_PER_BLOCK (16 * WAVES_PER_BLOCK)

// PRE: 0 = A as-is, 1 = A := (1+eps[layer])*A + A2   (GIN aggregation mix)
// ACT: 0 = none, 1 = relu, 2 = tanh
// D[M x ld] = act( pre(A)[M x K] @ B[K x ld] + bias[ld] )   (row-major)
// grid.x = ceil(M/128), grid.y = ld/64, block = 256 (8 waves)
// dynamic LDS = K*64*sizeof(float)
template <int PRE, int ACT>
__global__ void gemm_wmma_f32(const float* __restrict__ A,
                              const float* __restrict__ A2,
                              const float* __restrict__ epsArr, int layer,
                              const float* __restrict__ B,
                              const float* __restrict__ bias,
                              float* __restrict__ D,
                              int M, int K, int ld) {
  extern __shared__ float smem[];          // K/2 pairs x 64 cols x float2
  const int tid  = threadIdx.x;
  const int lane = tid & (WAVE - 1);
  const int wid  = tid >> 5;
  const int half = lane >> 4;              // 0: lanes 0-15, 1: lanes 16-31
  const int r    = lane & 15;
  const int n0   = blockIdx.y * 64;

  // ---- Cooperative fill of the B panel into LDS, paired along K ----
  // smem pair layout: pair p (K rows 2p,2p+1), col c -> smem[p*128 + c*2 + parity]
  {
    const int total = K * 64;
    for (int idx = tid; idx < total; idx += blockDim.x) {
      const int k = idx >> 6;              // K row
      const int c = idx & 63;              // panel col
      smem[(k >> 1) * 128 + c * 2 + (k & 1)] = B[(size_t)k * ld + n0 + c];
    }
  }
  __syncthreads();

  const int m0 = (blockIdx.x * WAVES_PER_BLOCK + wid) * 16;
  if (m0 < M) {                            // whole-wave guard (16 | M)
    float epsv = 1.0f;
    if constexpr (PRE == 1) epsv = 1.0f + epsArr[layer];

    v8f acc0 = {}, acc1 = {}, acc2 = {}, acc3 = {};
    const int arow = m0 + r;
    const float* __restrict__ Arow  = A + (size_t)arow * K;
    const float* __restrict__ A2row = (PRE == 1) ? (A2 + (size_t)arow * K) : nullptr;
    const float2* __restrict__ lds2 = (const float2*)smem;

    for (int kb = 0; kb < K; kb += 4) {
      const int k0 = kb + 2 * half;        // this half-wave's K pair
      // A fragment: 16x4 f32 (ISA 7.12.2): v0 = K even, v1 = K odd per half
      v2f a;
      if constexpr (PRE == 1) {
        a.x = epsv * Arow[k0]     + A2row[k0];
        a.y = epsv * Arow[k0 + 1] + A2row[k0 + 1];
      } else {
        a.x = Arow[k0];
        a.y = Arow[k0 + 1];
      }
      // B fragments from LDS: one ds_load_b64 each
      const int pbase = (k0 >> 1) * 64 + r;
      const float2 f0 = lds2[pbase];
      const float2 f1 = lds2[pbase + 16];
      const float2 f2 = lds2[pbase + 32];
      const float2 f3 = lds2[pbase + 48];
      v2f b0, b1, b2, b3;
      b0.x = f0.x; b0.y = f0.y;
      b1.x = f1.x; b1.y = f1.y;
      b2.x = f2.x; b2.y = f2.y;
      b3.x = f3.x; b3.y = f3.y;

      acc0 = __builtin_amdgcn_wmma_f32_16x16x4_f32(false, a, false, b0, (short)0, acc0, false, false);
      acc1 = __builtin_amdgcn_wmma_f32_16x16x4_f32(false, a, false, b1, (short)0, acc1, false, false);
      acc2 = __builtin_amdgcn_wmma_f32_16x16x4_f32(false, a, false, b2, (short)0, acc2, false, false);
      acc3 = __builtin_amdgcn_wmma_f32_16x16x4_f32(false, a, false, b3, (short)0, acc3, false, false);
    }

    // Epilogue. C/D layout: VGPR v holds M = m0+v (lanes 0-15) / m0+v+8 (16-31)
#pragma unroll
    for (int j = 0; j < 4; ++j) {
      const v8f acc = (j == 0) ? acc0 : (j == 1) ? acc1 : (j == 2) ? acc2 : acc3;
      const int col = n0 + 16 * j + r;
      const float bv = bias[col];
#pragma unroll
      for (int v = 0; v < 8; ++v) {
        const int row = m0 + v + 8 * half;
        float val = acc[v] + bv;
        if constexpr (ACT == 1) val = fmaxf(val, 0.0f);
        if constexpr (ACT == 2) val = tanhf(val);
        D[(size_t)row * ld + col] = val;
      }
    }
  }
}

// ---------------------------------------------------------------------------

__global__ void zero_kernel(float* __restrict__ p, size_t n) {
  size_t i = (size_t)blockIdx.x * blockDim.x + threadIdx.x;
  size_t stride = (size_t)gridDim.x * blockDim.x;
  for (; i < n; i += stride) p[i] = 0.0f;
}

// agg[dst[e]] += h[src[e]]   (float4 chunks, native global_atomic_add_f32)
__global__ void scatter_add_kernel(const float* __restrict__ h,
                                   const int* __restrict__ src,
                                   const int* __restrict__ dst,
                                   float* __restrict__ agg,
                                   int E, int H) {
  const int CH = H >> 2;  // float4 chunks per row
  size_t total = (size_t)E * CH;
  size_t i = (size_t)blockIdx.x * blockDim.x + threadIdx.x;
  size_t stride = (size_t)gridDim.x * blockDim.x;
  for (; i < total; i += stride) {
    const int e = (int)(i / CH);
    const int c = (int)(i % CH);
    const float4 v = ((const float4*)(h + (size_t)src[e] * H))[c];
    float* o = agg + (size_t)dst[e] * H + c * 4;
    __hip_atomic_fetch_add(o + 0, v.x, __ATOMIC_RELAXED, __HIP_MEMORY_SCOPE_AGENT);
    __hip_atomic_fetch_add(o + 1, v.y, __ATOMIC_RELAXED, __HIP_MEMORY_SCOPE_AGENT);
    __hip_atomic_fetch_add(o + 2, v.z, __ATOMIC_RELAXED, __HIP_MEMORY_SCOPE_AGENT);
    __hip_atomic_fetch_add(o + 3, v.w, __ATOMIC_RELAXED, __HIP_MEMORY_SCOPE_AGENT);
  }
}

// BatchNorm stats: block = H threads (thread = feature), coalesced row reads.
__global__ void bn_stats_kernel(const float* __restrict__ z,
                                float* __restrict__ sum,
                                float* __restrict__ sumsq,
                                int N, int H) {
  const int t = threadIdx.x;
  float s = 0.0f, s2 = 0.0f;
  for (int n = blockIdx.x; n < N; n += gridDim.x) {
    const float v = z[(size_t)n * H + t];
    s += v; s2 += v * v;
  }
  __hip_atomic_fetch_add(&sum[t],   s,  __ATOMIC_RELAXED, __HIP_MEMORY_SCOPE_AGENT);
  __hip_atomic_fetch_add(&sumsq[t], s2, __ATOMIC_RELAXED, __HIP_MEMORY_SCOPE_AGENT);
}

__global__ void bn_finalize_kernel(const float* __restrict__ sum,
                                   const float* __restrict__ sumsq,
                                   float* __restrict__ mu,
                                   float* __restrict__ rstd,
                                   int N) {
  const int t = threadIdx.x;
  const float invN = 1.0f / (float)N;
  const float m = sum[t] * invN;
  const float var = sumsq[t] * invN - m * m;
  mu[t] = m;
  rstd[t] = rsqrtf(var + 1e-5f);
}

// h = relu((z - mu) * rstd * gamma + beta)
__global__ void bn_apply_kernel(const float* __restrict__ z,
                                const float* __restrict__ mu,
                                const float* __restrict__ rstd,
                                const float* __restrict__ g,
                                const float* __restrict__ b,
                                float* __restrict__ h,
                                int N, int H) {
  const int t = threadIdx.x;
  const float m = mu[t], rs = rstd[t], gg = g[t], bb = b[t];
  for (int n = blockIdx.x; n < N; n += gridDim.x) {
    const float v = (z[(size_t)n * H + t] - m) * rs * gg + bb;
    h[(size_t)n * H + t] = fmaxf(v, 0.0f);
  }
}

// c[a] = sum_k h[seed, k] * W2[k, a]
__global__ void seed_vec_kernel(const float* __restrict__ h,
                                const int* __restrict__ seedp,
                                const float* __restrict__ W2,
                                float* __restrict__ c,
                                int H, int A) {
  const int a = threadIdx.x;
  if (a >= A) return;
  const int s = seedp[0];
  const float* hs = h + (size_t)s * H;
  float acc = 0.0f;
  for (int k = 0; k < H; ++k) acc += hs[k] * W2[(size_t)k * A + a];
  c[a] = acc;
}

// e[n] = leaky_relu( zat[n,:] @ w_attn , 0.2 )  — one wave per row
__global__ void attn_e_kernel(const float* __restrict__ zat,
                              const float* __restrict__ w_attn,
                              float* __restrict__ e,
                              int N, int A) {
  const int lane = threadIdx.x & (WAVE - 1);
  const int wid  = threadIdx.x >> 5;
  const int wpb  = blockDim.x >> 5;
  for (int row = blockIdx.x * wpb + wid; row < N; row += gridDim.x * wpb) {
    float acc = 0.0f;
    const float* zr = zat + (size_t)row * A;
    for (int j = lane; j < A; j += WAVE) acc += zr[j] * w_attn[j];
#pragma unroll
    for (int off = 16; off > 0; off >>= 1) acc += __shfl_xor(acc, off, WAVE);
    if (lane == 0) e[row] = (acc > 0.0f) ? acc : 0.2f * acc;
  }
}

__global__ void reduce_max_kernel(const float* __restrict__ e,
                                  float* __restrict__ out, int N) {
  __shared__ float red[1024];
  const int t = threadIdx.x;
  float m = -3.4e38f;
  for (int i = t; i < N; i += blockDim.x) m = fmaxf(m, e[i]);
  red[t] = m;
  __syncthreads();
  for (int s = blockDim.x >> 1; s > 0; s >>= 1) {
    if (t < s) red[t] = fmaxf(red[t], red[t + s]);
    __syncthreads();
  }
  if (t == 0) out[0] = red[0];
}

__global__ void reduce_sumexp_kernel(const float* __restrict__ e,
                                     const float* __restrict__ mx,
                                     float* __restrict__ out, int N) {
  __shared__ float red[1024];
  const int t = threadIdx.x;
  const float m = mx[0];
  float s = 0.0f;
  for (int i = t; i < N; i += blockDim.x) s += expf(e[i] - m);
  red[t] = s;
  __syncthreads();
  for (int st = blockDim.x >> 1; st > 0; st >>= 1) {
    if (t < st) red[t] += red[t + st];
    __syncthreads();
  }
  if (t == 0) out[0] = red[0];
}

__global__ void alpha_kernel(const float* __restrict__ e,
                             const float* __restrict__ mx,
                             const float* __restrict__ se,
                             float* __restrict__ alpha, int N) {
  const float m = mx[0];
  const float inv = 1.0f / se[0];
  int i = blockIdx.x * blockDim.x + threadIdx.x;
  const int stride = gridDim.x * blockDim.x;
  for (; i < N; i += stride) alpha[i] = expf(e[i] - m) * inv;
}

// readout[f] += sum_n alpha[n] * h[n,f]  (block = H threads, thread = feature)
__global__ void readout_kernel(const float* __restrict__ alpha,
                               const float* __restrict__ h,
                               float* __restrict__ readout,
                               int N, int H) {
  const int t = threadIdx.x;
  float acc = 0.0f;
  for (int n = blockIdx.x; n < N; n += gridDim.x)
    acc += alpha[n] * h[(size_t)n * H + t];
  __hip_atomic_fetch_add(&readout[t], acc, __ATOMIC_RELAXED, __HIP_MEMORY_SCOPE_AGENT);
}

__global__ void logit_kernel(const float* __restrict__ readout,
                             const float* __restrict__ cls_w,
                             const float* __restrict__ cls_b,
                             float* __restrict__ out, int H) {
  __shared__ float red[256];
  const int t = threadIdx.x;
  red[t] = (t < H) ? readout[t] * cls_w[t] : 0.0f;
  __syncthreads();
  for (int s = blockDim.x >> 1; s > 0; s >>= 1) {
    if (t < s) red[t] += red[t + s];
    __syncthreads();
  }
  if (t == 0) out[0] = red[0] + cls_b[0];
}

// ---------------------------------------------------------------------------

extern "C" void kernel_launch(void* const* d_in, const int* in_sizes, int n_in,
                              void* d_out, int out_size, void* d_ws, size_t ws_size,
                              hipStream_t stream) {
  const float* x       = (const float*)d_in[0];
  const int*   eidx    = (const int*)d_in[1];
  const int*   seedp   = (const int*)d_in[2];
  const float* proj_w  = (const float*)d_in[3];
  const float* proj_b  = (const float*)d_in[4];
  const float* mlp1_w  = (const float*)d_in[5];
  const float* mlp1_b  = (const float*)d_in[6];
  const float* mlp2_w  = (const float*)d_in[7];
  const float* mlp2_b  = (const float*)d_in[8];
  const float* epsArr  = (const float*)d_in[9];
  const float* bn_g    = (const float*)d_in[10];
  const float* bn_b    = (const float*)d_in[11];
  const float* W1      = (const float*)d_in[12];
  const float* W2      = (const float*)d_in[13];
  const float* w_attn  = (const float*)d_in[14];
  const float* cls_w   = (const float*)d_in[15];
  const float* cls_b   = (const float*)d_in[16];

  const int H  = in_sizes[4];             // 192
  const int IN = in_sizes[3] / H;         // 128
  const int N  = in_sizes[0] / IN;        // 50000 (divisible by 16)
  const int E  = in_sizes[1] / 2;         // 400000
  const int L  = in_sizes[9];             // 3
  const int A  = in_sizes[14];            // 192

  const int* src = eidx;
  const int* dst = eidx + E;

  float* ws   = (float*)d_ws;
  const size_t NH = (size_t)N * H;
  float* h    = ws;                 // [N,H]
  float* bufA = ws + NH;            // agg / z / zat
  float* bufB = ws + 2 * NH;        // mlp1 intermediate / e
  float* sm   = ws + 3 * NH;        // small scalars
  float* sum   = sm;
  float* sumsq = sm + H;
  float* mu    = sm + 2 * H;
  float* rstd  = sm + 3 * H;
  float* cvec  = sm + 4 * H;        // size A
  float* rdout = sm + 4 * H + A;    // size H
  float* red   = sm + 5 * H + A;    // [0]=max, [1]=sumexp

  float* logit = (float*)d_out;
  float* alpha = (float*)d_out + 1;

  const int mBlocks = (N + ROWS_PER_BLOCK - 1) / ROWS_PER_BLOCK;
  const dim3 gH(mBlocks, H / 64);   // GEMM grids (output ld = H)
  const dim3 gA(mBlocks, A / 64);   // (output ld = A)
  const size_t ldsIN = (size_t)IN * 64 * sizeof(float);  // 32 KB
  const size_t ldsH  = (size_t)H  * 64 * sizeof(float);  // 48 KB
  const int rowsGrid = 512;
  const int BT = WAVE * WAVES_PER_BLOCK;  // 256 threads

  // h = relu(x @ proj_w + proj_b)
  gemm_wmma_f32<0, 1><<<gH, BT, ldsIN, stream>>>(x, nullptr, nullptr, 0,
                                                 proj_w, proj_b, h, N, IN, H);

  for (int l = 0; l < L; ++l) {
    // agg = segment_sum(h[src], dst)
    zero_kernel<<<4096, 256, 0, stream>>>(bufA, NH);
    {
      const size_t total = (size_t)E * (H / 4);
      int blocks = (int)((total + 255) / 256);
      if (blocks > 65535) blocks = 65535;
      scatter_add_kernel<<<blocks, 256, 0, stream>>>(h, src, dst, bufA, E, H);
    }
    // t = relu(((1+eps)h + agg) @ mlp1_w[l] + mlp1_b[l])
    gemm_wmma_f32<1, 1><<<gH, BT, ldsH, stream>>>(
        h, bufA, epsArr, l, mlp1_w + (size_t)l * H * H, mlp1_b + (size_t)l * H,
        bufB, N, H, H);
    // z = t @ mlp2_w[l] + mlp2_b[l]
    gemm_wmma_f32<0, 0><<<gH, BT, ldsH, stream>>>(
        bufB, nullptr, nullptr, 0, mlp2_w + (size_t)l * H * H,
        mlp2_b + (size_t)l * H, bufA, N, H, H);
    // BatchNorm (batch statistics) + relu -> h
    zero_kernel<<<2, 256, 0, stream>>>(sum, 2 * (size_t)H);
    bn_stats_kernel<<<rowsGrid, H, 0, stream>>>(bufA, sum, sumsq, N, H);
    bn_finalize_kernel<<<1, H, 0, stream>>>(sum, sumsq, mu, rstd, N);
    bn_apply_kernel<<<rowsGrid, H, 0, stream>>>(bufA, mu, rstd,
                                                bn_g + (size_t)l * H,
                                                bn_b + (size_t)l * H, h, N, H);
  }

  // Seed attention
  seed_vec_kernel<<<1, A, 0, stream>>>(h, seedp, W2, cvec, H, A);
  // zat = tanh(h @ W1 + c)  (c plays the per-column bias role)
  gemm_wmma_f32<0, 2><<<gA, BT, ldsH, stream>>>(h, nullptr, nullptr, 0,
                                                W1, cvec, bufA, N, H, A);
  // e = leaky_relu(zat @ w_attn, 0.2)
  attn_e_kernel<<<2048, 256, 0, stream>>>(bufA, w_attn, bufB, N, A);
  // softmax over nodes
  reduce_max_kernel<<<1, 1024, 0, stream>>>(bufB, red, N);
  reduce_sumexp_kernel<<<1, 1024, 0, stream>>>(bufB, red, red + 1, N);
  alpha_kernel<<<256, 256, 0, stream>>>(bufB, red, red + 1, alpha, N);
  // readout = alpha @ h ; logit = readout @ cls_w + cls_b
  zero_kernel<<<1, 256, 0, stream>>>(rdout, (size_t)H);
  readout_kernel<<<rowsGrid, H, 0, stream>>>(alpha, h, rdout, N, H);
  logit_kernel<<<1, 256, 0, stream>>>(rdout, cls_w, cls_b, logit, H);
}